// Field_34325378630169
// MI455X (gfx1250) — compile-verified
//
#include <hip/hip_runtime.h>
#include <math.h>

// ---------------------------------------------------------------------------
// VolSDF-style volume renderer for MI455X (gfx1250, wave32).
//
// Transmittance cumprod is computed as a matmul prefix-scan on the WMMA unit:
//   1 - alpha = exp(-sigma*dt)  =>  T_excl[n,s] = exp(-sum_{k<s} sigma[n,k]*dt)
//   cum(16x64) = A(16x64) x L(64x64), L strictly-lower-triangular ones,
// chained as V_WMMA_F32_16X16X4_F32 (full fp32 accumulate, no precision loss).
// ---------------------------------------------------------------------------

typedef float v2f __attribute__((ext_vector_type(2)));
typedef float v8f __attribute__((ext_vector_type(8)));

constexpr int   kR       = 192;
constexpr float kVoxel   = 3.0f / 192.0f;
constexpr float kInvVox  = 192.0f / 3.0f;
constexpr float kOrigin  = -1.5f;
constexpr float kNear    = 0.2f;
constexpr float kFar     = 1.8f;
constexpr float kDt      = (kFar - kNear) / 64.0f;   // 0.025
constexpr float kMinBeta = kVoxel;

__device__ __forceinline__ float lerpf(float a, float b, float t) {
  return fmaf(t, b - a, a);
}

__device__ __forceinline__ float4 lerp4(const float4& a, const float4& b, float t) {
  return make_float4(fmaf(t, b.x - a.x, a.x), fmaf(t, b.y - a.y, a.y),
                     fmaf(t, b.z - a.z, a.z), fmaf(t, b.w - a.w, a.w));
}

__device__ __forceinline__ int cell_index(int x, int y, int z) {
  return (x * kR + y) * kR + z;    // max 192^3-1 ~ 7.1M, fits int32
}

// Laplace-CDF density (VolSDF): sigma = (0.5 + 0.5*sign(s)*expm1(-|s|/b)) / b
__device__ __forceinline__ float laplace_sigma(float sdf, float inv_beta) {
  float em = expm1f(-fabsf(sdf) * inv_beta);
  float sg = (sdf > 0.0f) ? 1.0f : ((sdf < 0.0f) ? -1.0f : 0.0f);
  return inv_beta * fmaf(0.5f * sg, em, 0.5f);
}

// sdf-channel-only trilinear fetch (8x b32 gathers), returns in-bounds mask too
__device__ __forceinline__ float tri_sdf(const float* __restrict__ grid,
                                         float px, float py, float pz,
                                         float* inb_out) {
  float ux = (px - kOrigin) * kInvVox;
  float uy = (py - kOrigin) * kInvVox;
  float uz = (pz - kOrigin) * kInvVox;
  float inb = (ux >= 0.0f && ux <= (float)(kR - 1) &&
               uy >= 0.0f && uy <= (float)(kR - 1) &&
               uz >= 0.0f && uz <= (float)(kR - 1)) ? 1.0f : 0.0f;
  float cx = fminf(fmaxf(ux, 0.0f), (float)(kR - 1) - 1e-5f);
  float cy = fminf(fmaxf(uy, 0.0f), (float)(kR - 1) - 1e-5f);
  float cz = fminf(fmaxf(uz, 0.0f), (float)(kR - 1) - 1e-5f);
  int x0 = (int)cx, y0 = (int)cy, z0 = (int)cz;
  float fx = cx - (float)x0, fy = cy - (float)y0, fz = cz - (float)z0;
  int x1 = x0 + 1, y1 = y0 + 1, z1 = z0 + 1;

  float c000 = grid[4 * cell_index(x0, y0, z0)];
  float c100 = grid[4 * cell_index(x1, y0, z0)];
  float c010 = grid[4 * cell_index(x0, y1, z0)];
  float c110 = grid[4 * cell_index(x1, y1, z0)];
  float c001 = grid[4 * cell_index(x0, y0, z1)];
  float c101 = grid[4 * cell_index(x1, y0, z1)];
  float c011 = grid[4 * cell_index(x0, y1, z1)];
  float c111 = grid[4 * cell_index(x1, y1, z1)];

  float v00 = lerpf(c000, c100, fx);
  float v10 = lerpf(c010, c110, fx);
  float v01 = lerpf(c001, c101, fx);
  float v11 = lerpf(c011, c111, fx);
  float v0  = lerpf(v00, v10, fy);
  float v1  = lerpf(v01, v11, fy);
  *inb_out = inb;
  return lerpf(v0, v1, fz);
}

// One wave handles 16 rays x 64 samples. 4 waves / block => 64 rays / block.
__global__ __launch_bounds__(128, 1)
void volsdf_wmma_kernel(const float* __restrict__ rays_o,
                        const float* __restrict__ rays_d,
                        const float* __restrict__ rays_d_norm,
                        const float* __restrict__ grid,
                        const float* __restrict__ beta,
                        float* __restrict__ out) {
  // cum[ray 0..15][sample 0..63], padded stride 68 -> conflict-free banks
  __shared__ float cum_lds[4][16][68];

  const int tid  = threadIdx.x;
  const int wave = tid >> 5;
  const int lane = tid & 31;
  const int mrow = lane & 15;   // ray row within wave (A/B/C lane mapping)
  const int half = lane >> 4;   // which half-wave

  const int ray = blockIdx.x * 64 + wave * 16 + mrow;

  const float ox = rays_o[3 * ray + 0];
  const float oy = rays_o[3 * ray + 1];
  const float oz = rays_o[3 * ray + 2];
  const float dx = rays_d[3 * ray + 0];
  const float dy = rays_d[3 * ray + 1];
  const float dz = rays_d[3 * ray + 2];

  const float beta_eff = kMinBeta + fabsf(beta[0]);
  const float inv_beta = 1.0f / beta_eff;

  // ------------------------------------------------------------------
  // Pass 1: sigma*dt directly in WMMA A-fragment layout.
  // A (16x4 f32) chunk c: lanes 0-15 hold K=4c+{0,1}, lanes 16-31 K=4c+{2,3}.
  // ------------------------------------------------------------------
  v2f afrag[16];
  #pragma unroll
  for (int c = 0; c < 16; ++c) {
    #pragma unroll
    for (int e = 0; e < 2; ++e) {
      int s = 4 * c + 2 * half + e;
      float tm = kNear + kDt * ((float)s + 0.5f);
      float inb;
      float sdf = tri_sdf(grid, fmaf(tm, dx, ox), fmaf(tm, dy, oy),
                          fmaf(tm, dz, oz), &inb);
      float sigma = laplace_sigma(sdf, inv_beta) * inb;
      if (e == 0) afrag[c].x = sigma * kDt;
      else        afrag[c].y = sigma * kDt;
    }
  }

  // ------------------------------------------------------------------
  // WMMA prefix-scan: cum[m][16j+n] = sum_{k < 16j+n} sigma*dt.
  // B (4x16) built in registers: B[k][n] = (k_global < 16j + n).
  // Only chunks at/below each output block's diagonal contribute: 40 WMMAs.
  // ------------------------------------------------------------------
  v8f cum[4];
  #pragma unroll
  for (int j = 0; j < 4; ++j) {
    v8f acc;
    #pragma unroll
    for (int q = 0; q < 8; ++q) acc[q] = 0.0f;
    #pragma unroll
    for (int c = 0; c < 16; ++c) {
      if (c <= 4 * j + 3) {             // compile-time pruned
        int kg = 4 * c + 2 * half;      // global k of this lane's element 0
        v2f b;
        b.x = (kg     < 16 * j + mrow) ? 1.0f : 0.0f;
        b.y = (kg + 1 < 16 * j + mrow) ? 1.0f : 0.0f;
        acc = __builtin_amdgcn_wmma_f32_16x16x4_f32(
            /*neg_a=*/false, afrag[c], /*neg_b=*/false, b,
            /*c_mod=*/(short)0, acc, /*reuse_a=*/false, /*reuse_b=*/false);
      }
    }
    cum[j] = acc;
  }

  // D layout -> LDS: VGPR v of lane L is cum[row v+8*half][col 16j + (L&15)]
  #pragma unroll
  for (int j = 0; j < 4; ++j) {
    #pragma unroll
    for (int v = 0; v < 8; ++v) {
      cum_lds[wave][v + 8 * half][16 * j + mrow] = cum[j][v];
    }
  }
  __syncthreads();

  // ------------------------------------------------------------------
  // Pass 2: full shading. Lane L = ray (L&15), samples sb + (L>>4).
  // 8x float4 (global_load_b128) gathers per sample; analytic sdf gradient.
  // ------------------------------------------------------------------
  float a_r = 0.0f, a_g = 0.0f, a_b = 0.0f, a_d = 0.0f;
  float a_nx = 0.0f, a_ny = 0.0f, a_nz = 0.0f, a_w = 0.0f;

  const float4* __restrict__ g4 = (const float4*)grid;

  for (int sb = 0; sb < 64; sb += 2) {
    int s = sb + half;
    float tm = kNear + kDt * ((float)s + 0.5f);
    float px = fmaf(tm, dx, ox);
    float py = fmaf(tm, dy, oy);
    float pz = fmaf(tm, dz, oz);

    float ux = (px - kOrigin) * kInvVox;
    float uy = (py - kOrigin) * kInvVox;
    float uz = (pz - kOrigin) * kInvVox;
    float inb = (ux >= 0.0f && ux <= (float)(kR - 1) &&
                 uy >= 0.0f && uy <= (float)(kR - 1) &&
                 uz >= 0.0f && uz <= (float)(kR - 1)) ? 1.0f : 0.0f;
    float cx = fminf(fmaxf(ux, 0.0f), (float)(kR - 1) - 1e-5f);
    float cy = fminf(fmaxf(uy, 0.0f), (float)(kR - 1) - 1e-5f);
    float cz = fminf(fmaxf(uz, 0.0f), (float)(kR - 1) - 1e-5f);
    int x0 = (int)cx, y0 = (int)cy, z0 = (int)cz;
    float fx = cx - (float)x0, fy = cy - (float)y0, fz = cz - (float)z0;
    int x1 = x0 + 1, y1 = y0 + 1, z1 = z0 + 1;

    float4 c000 = g4[cell_index(x0, y0, z0)];
    float4 c100 = g4[cell_index(x1, y0, z0)];
    float4 c010 = g4[cell_index(x0, y1, z0)];
    float4 c110 = g4[cell_index(x1, y1, z0)];
    float4 c001 = g4[cell_index(x0, y0, z1)];
    float4 c101 = g4[cell_index(x1, y0, z1)];
    float4 c011 = g4[cell_index(x0, y1, z1)];
    float4 c111 = g4[cell_index(x1, y1, z1)];

    // trilinear value (all 4 channels)
    float4 v00 = lerp4(c000, c100, fx);
    float4 v10 = lerp4(c010, c110, fx);
    float4 v01 = lerp4(c001, c101, fx);
    float4 v11 = lerp4(c011, c111, fx);
    float4 v0  = lerp4(v00, v10, fy);
    float4 v1  = lerp4(v01, v11, fy);
    float4 val = lerp4(v0, v1, fz);

    // analytic sdf gradient (channel .x), d/dx = d/du * (1/VOXEL)
    float dxa = lerpf(c100.x - c000.x, c110.x - c010.x, fy);
    float dxb = lerpf(c101.x - c001.x, c111.x - c011.x, fy);
    float gx  = lerpf(dxa, dxb, fz) * kInvVox;
    float dya = lerpf(c010.x - c000.x, c110.x - c100.x, fx);
    float dyb = lerpf(c011.x - c001.x, c111.x - c101.x, fx);
    float gy  = lerpf(dya, dyb, fz) * kInvVox;
    float dza = lerpf(c001.x - c000.x, c101.x - c100.x, fx);
    float dzb = lerpf(c011.x - c010.x, c111.x - c110.x, fx);
    float gz  = lerpf(dza, dzb, fy) * kInvVox;

    float invn = 1.0f / fmaxf(sqrtf(gx * gx + gy * gy + gz * gz), 1e-12f);

    float sigma = laplace_sigma(val.x, inv_beta) * inb;
    float cumv  = cum_lds[wave][mrow][s];
    float T     = expf(-cumv);                 // exclusive transmittance
    float alpha = -expm1f(-sigma * kDt);       // 1 - exp(-sigma*dt), exact
    float w     = alpha * T;

    a_r  += w * val.y;
    a_g  += w * val.z;
    a_b  += w * val.w;
    a_d  += w * tm;
    a_nx += w * gx * invn;
    a_ny += w * gy * invn;
    a_nz += w * gz * invn;
    a_w  += w;
  }

  // combine even/odd sample halves (lane L <-> L^16)
  a_r  += __shfl_xor(a_r, 16, 32);
  a_g  += __shfl_xor(a_g, 16, 32);
  a_b  += __shfl_xor(a_b, 16, 32);
  a_d  += __shfl_xor(a_d, 16, 32);
  a_nx += __shfl_xor(a_nx, 16, 32);
  a_ny += __shfl_xor(a_ny, 16, 32);
  a_nz += __shfl_xor(a_nz, 16, 32);
  a_w  += __shfl_xor(a_w, 16, 32);

  if (half == 0) {
    float dn = rays_d_norm[ray];
    float4* o4 = (float4*)out;                 // [rgb, depth][normal, acc]
    o4[ray * 2 + 0] = make_float4(a_r, a_g, a_b, a_d / dn);
    o4[ray * 2 + 1] = make_float4(a_nx, a_ny, a_nz, a_w);
  }
}

extern "C" void kernel_launch(void* const* d_in, const int* in_sizes, int n_in,
                              void* d_out, int out_size, void* d_ws, size_t ws_size,
                              hipStream_t stream) {
  (void)n_in; (void)out_size; (void)d_ws; (void)ws_size;
  const float* rays_o      = (const float*)d_in[0];
  const float* rays_d      = (const float*)d_in[1];
  const float* rays_d_norm = (const float*)d_in[2];
  const float* grid        = (const float*)d_in[3];
  const float* beta        = (const float*)d_in[4];
  float* out = (float*)d_out;

  int nrays  = in_sizes[0] / 3;   // 32768
  int blocks = nrays / 64;        // 64 rays per 128-thread block (4 waves)

  hipLaunchKernelGGL(volsdf_wmma_kernel, dim3(blocks), dim3(128), 0, stream,
                     rays_o, rays_d, rays_d_norm, grid, beta, out);
}